// HNN_74259984548160
// MI455X (gfx1250) — compile-verified
//
#include <hip/hip_runtime.h>
#include <math.h>

// ---------------------------------------------------------------------------
// HGT forward for MI455X (gfx1250, wave32).
// Dense projections -> V_WMMA_F32_16X16X4_F32 (exact fp32, matches reference),
// 8 waves/block, W panel staged in LDS (transposed -> ds_load b64 B-frags),
// NT=4 N-tile unroll per wave; B-frags batched ahead of the WMMA group so the
// LDS loads overlap instead of serializing on dscnt.
// Edge softmax/aggregation -> 3 passes with global f32 atomics.
// ---------------------------------------------------------------------------

typedef __attribute__((ext_vector_type(2))) float v2f;
typedef __attribute__((ext_vector_type(8))) float v8f;

#define HID 64
#define NHEAD 4
#define DH 16
#define NTOT_NODES 225000
#define WAVES_PER_BLOCK 8

__device__ __forceinline__ float gelu_exact(float x) {
    return 0.5f * x * (1.0f + erff(x * 0.70710678118654752f));
}

__device__ __forceinline__ void atomic_max_f32(float* addr, float val) {
    if (val >= 0.f) atomicMax((int*)addr, __float_as_int(val));
    else            atomicMin((unsigned int*)addr, __float_as_uint(val));
}

__global__ void fill_f32(float* p, float v, long long n) {
    long long i = (long long)blockIdx.x * blockDim.x + threadIdx.x;
    long long stride = (long long)gridDim.x * blockDim.x;
    for (; i < n; i += stride) p[i] = v;
}

__global__ void gelu_fwd(const float* __restrict__ in, float* __restrict__ out, long long n) {
    long long i = (long long)blockIdx.x * blockDim.x + threadIdx.x;
    long long stride = (long long)gridDim.x * blockDim.x;
    for (; i < n; i += stride) out[i] = gelu_exact(in[i]);
}

// ---------------------------------------------------------------------------
// fp32 WMMA GEMM:  C[M,Ncols] = epi( A[M,K] @ W[K,Ncols] + bias )
//   blockDim = 256 (8 waves); wave w -> rows m0 = (bx*8+w)*16
//   blockIdx.y -> group of NT 16-wide column tiles; blockIdx.z -> "head"
//   (per-head element offsets hA/hW/hC; 0-stride for plain GEMMs).
// W panel ([K, NT*16], K<=64) is cooperatively staged to LDS transposed
// ([col][k]) so each B fragment is a contiguous 64-bit LDS load; the panel
// is shared by all 8 waves. Index clamping (not branching) handles M/Ncols
// edges: clamped lanes feed garbage only into rows/cols never stored.
// ---------------------------------------------------------------------------
template<int NT>
__global__ __launch_bounds__(256) void gemm_wmma(
    const float* __restrict__ A, int lda,
    const float* __restrict__ W, int ldw,
    const float* __restrict__ bias,
    float* __restrict__ C, int ldc,
    int M, int K, int Ncols,
    long hA, long hW, long hC,
    int relu_out,
    const float* __restrict__ skip_old,     // same ld as C when used
    const float* __restrict__ skip_param)   // device scalar (pre-sigmoid)
{
    __shared__ float ldsw[64 * NT * 16];    // [col][k], supports K <= 64
    const int NTC = NT * 16;
    const int tid  = threadIdx.x;
    const int lane = tid & 31;
    const int wave = tid >> 5;

    A += (long)blockIdx.z * hA;
    W += (long)blockIdx.z * hW;
    C += (long)blockIdx.z * hC;
    const float* skp = skip_old ? (skip_old + (long)blockIdx.z * hC) : nullptr;

    const int n0 = blockIdx.y * NTC;

    // stage W panel -> LDS (transposed); clamp columns at the N edge
    for (int i = tid; i < NTC * K; i += 256) {
        int col = i % NTC;
        int k   = i / NTC;
        int gcol = n0 + col; if (gcol > Ncols - 1) gcol = Ncols - 1;
        ldsw[col * K + k] = W[(size_t)k * ldw + gcol];
    }
    __syncthreads();

    const int m0 = (blockIdx.x * WAVES_PER_BLOCK + wave) * 16;
    if (m0 >= M) return;

    const int r16 = lane & 15;
    const int kh  = (lane >> 4) * 2;
    int arow = m0 + r16; if (arow > M - 1) arow = M - 1;   // clamp, no branch
    const float* ap = A + (size_t)arow * lda;

    v8f zero = {};
    v8f acc[NT];
#pragma unroll
    for (int t = 0; t < NT; ++t) acc[t] = zero;

    for (int k0 = 0; k0 < K; k0 += 4) {     // K is always a multiple of 4
        v2f a;
        a.x = ap[k0 + kh];
        a.y = ap[k0 + kh + 1];
        // batch all NT B-fragment loads (independent regs -> overlapped ds ops)
        v2f b[NT];
#pragma unroll
        for (int t = 0; t < NT; ++t) {
            const float* wp = &ldsw[(t * 16 + r16) * K + (k0 + kh)];
            b[t].x = wp[0];
            b[t].y = wp[1];
        }
#pragma unroll
        for (int t = 0; t < NT; ++t)
            acc[t] = __builtin_amdgcn_wmma_f32_16x16x4_f32(false, a, false, b[t],
                                                           (short)0, acc[t], false, false);
    }

    const int mh = (lane >> 4) * 8;
    float blend = 0.f;
    if (skp) blend = 1.f / (1.f + expf(-skip_param[0]));
#pragma unroll
    for (int t = 0; t < NT; ++t) {
        int col = n0 + t * 16 + r16;
        if (col >= Ncols) continue;
        float bi = bias ? bias[col] : 0.f;
#pragma unroll
        for (int i = 0; i < 8; ++i) {
            int row = m0 + mh + i;
            if (row < M) {
                float v = acc[t][i] + bi;
                if (relu_out) v = fmaxf(v, 0.f);
                if (skp) v = blend * v + (1.f - blend) * skp[(size_t)row * ldc + col];
                C[(size_t)row * ldc + col] = v;
            }
        }
    }
}

// ---------------------------------------------------------------------------
// Edge pass 1: attention logit + segment max (one thread per edge*head)
// ---------------------------------------------------------------------------
__global__ void edge_logits(const int* __restrict__ src, const int* __restrict__ dst, int E,
                            const float* __restrict__ kt,     // [Ns, 64]
                            const float* __restrict__ kqv_d,  // dest-type fused KQV [Nd,192]
                            const float* __restrict__ p_rel,  // [4]
                            float* __restrict__ logit_out,    // [E, 4] slice
                            float* __restrict__ mmax,         // [NTOT, 4]
                            int off_d)
{
    int idx = blockIdx.x * blockDim.x + threadIdx.x;
    int e = idx >> 2, h = idx & 3;
    if (e >= E) return;
    int s = src[e], d = dst[e];
    const float* q = kqv_d + (size_t)d * 192 + 64 + h * DH;   // Q = cols 64..127
    const float* k = kt + (size_t)s * HID + h * DH;
    float acc = 0.f;
#pragma unroll
    for (int i = 0; i < DH; ++i) acc += q[i] * k[i];
    float lg = acc * p_rel[h] * 0.25f;                        // 1/sqrt(16)
    logit_out[(size_t)e * 4 + h] = lg;
    atomic_max_f32(&mmax[(size_t)(off_d + d) * 4 + h], lg);
}

// ---------------------------------------------------------------------------
// Edge pass 2: exp(logit - m) in place + denominator atomicAdd
// ---------------------------------------------------------------------------
__global__ void edge_exp_den(const int* __restrict__ dst, int E,
                             float* __restrict__ logit_io,
                             const float* __restrict__ mmax,
                             float* __restrict__ den, int off_d)
{
    int idx = blockIdx.x * blockDim.x + threadIdx.x;
    int e = idx >> 2, h = idx & 3;
    if (e >= E) return;
    int d = dst[e];
    size_t gi = (size_t)(off_d + d) * 4 + h;
    float ex = expf(logit_io[(size_t)e * 4 + h] - mmax[gi]);
    logit_io[(size_t)e * 4 + h] = ex;
    atomicAdd(&den[gi], ex);
}

// ---------------------------------------------------------------------------
// Edge pass 3: alpha-weighted scatter of relation-transformed V
// ---------------------------------------------------------------------------
__global__ void edge_aggregate(const int* __restrict__ src, const int* __restrict__ dst, int E,
                               const float* __restrict__ ex,
                               const float* __restrict__ den,
                               const float* __restrict__ vt,   // [Ns,64]
                               float* __restrict__ agg,        // [NTOT,64]
                               int off_d)
{
    int idx = blockIdx.x * blockDim.x + threadIdx.x;
    int e = idx >> 2, h = idx & 3;
    if (e >= E) return;
    int s = src[e], d = dst[e];
    size_t gi = (size_t)(off_d + d) * 4 + h;
    float alpha = ex[(size_t)e * 4 + h] / (den[gi] + 1e-16f);
    const float* v = vt + (size_t)s * HID + h * DH;
    float* o = agg + (size_t)(off_d + d) * HID + h * DH;
#pragma unroll
    for (int i = 0; i < DH; ++i) atomicAdd(&o[i], alpha * v[i]);
}

// ---------------------------------------------------------------------------
// Host orchestration (all launches on `stream`; graph-capture safe)
// ---------------------------------------------------------------------------
extern "C" void kernel_launch(void* const* d_in, const int* in_sizes, int n_in,
                              void* d_out, int out_size, void* d_ws, size_t ws_size,
                              hipStream_t stream)
{
    (void)in_sizes; (void)n_in; (void)out_size; (void)ws_size;

    const float* x_in[3] = {(const float*)d_in[0], (const float*)d_in[1], (const float*)d_in[2]};
    const int* edg[6];
    for (int i = 0; i < 6; ++i) edg[i] = (const int*)d_in[3 + i];

    // node meta: 0=player, 1=team, 2=match
    static const int NN[3]   = {100000, 5000, 120000};
    static const int OFFN[3] = {0, 100000, 105000};
    static const int FIN[3]  = {64, 32, 48};
    static const int SPOS[3] = {1, 2, 0};   // pos of (player,team,match) among {match,player,team}

    // params leaf layout (jax pytree: dict keys sorted alphabetically)
    const int CONV0 = 9, CONV_STRIDE = 33;
    const int LIN_IN = 75, LIN_OUT_B = 81, LIN_OUT_W = 82;

    // edge meta in EDGE_TYPES order; sidx = alphabetical index of edge-type name
    struct EM { int ei; int E; int offs; int offd; int Ns; int sidx; long eoff; };
    const EM em[6] = {
        {0, 200000, 0,      100000, 100000, 3, 0},        // player->plays_for->team
        {1, 200000, 100000, 0,      5000,   4, 200000},   // team->has->player
        {2, 600000, 0,      105000, 100000, 2, 400000},   // player->in->match
        {3, 600000, 105000, 0,      120000, 0, 1000000},  // match->contains->player
        {4, 200000, 100000, 105000, 5000,   5, 1600000},  // team->played->match
        {5, 200000, 105000, 100000, 120000, 1, 1800000},  // match->of->team
    };
    const long ETOT = 2000000;

    // workspace carve (floats)
    float* w = (float*)d_ws;
    float* x_a    = w; w += (size_t)NTOT_NODES * HID;
    float* x_b    = w; w += (size_t)NTOT_NODES * HID;
    float* kqv    = w; w += (size_t)NTOT_NODES * 192;
    float* trans  = w; w += (size_t)120000 * HID;        // per-edge-type K/V relation scratch
    float* logits = w; w += (size_t)ETOT * 4;
    float* mmax   = w; w += (size_t)NTOT_NODES * 4;
    float* den    = w; w += (size_t)NTOT_NODES * 4;
    float* agg    = w; w += (size_t)NTOT_NODES * HID;
    float* gact   = w; w += (size_t)NTOT_NODES * HID;    // gelu(agg)

    const int MROWS = WAVES_PER_BLOCK * 16;              // rows per block

    // NT=4 GEMM, no head strides
    auto gemm4 = [&](const float* A, int lda, const float* Wm, int ldw, const float* bias,
                     float* C, int ldc, int M, int K, int Ncols,
                     int relu_out, const float* skip_old, const float* skip_param) {
        dim3 g((unsigned)((M + MROWS - 1) / MROWS), (unsigned)((Ncols + 63) / 64), 1);
        hipLaunchKernelGGL((gemm_wmma<4>), g, dim3(256), 0, stream,
                           A, lda, Wm, ldw, bias, C, ldc, M, K, Ncols,
                           0L, 0L, 0L, relu_out, skip_old, skip_param);
    };
    // NT=1 GEMM with per-head strides (blockIdx.z = head)
    auto gemm1h = [&](const float* A, int lda, const float* Wm, int ldw, const float* bias,
                      float* C, int ldc, int M, int K, int Ncols,
                      long hA, long hW, long hC, int heads) {
        dim3 g((unsigned)((M + MROWS - 1) / MROWS), (unsigned)((Ncols + 15) / 16), (unsigned)heads);
        hipLaunchKernelGGL((gemm_wmma<1>), g, dim3(256), 0, stream,
                           A, lda, Wm, ldw, bias, C, ldc, M, K, Ncols,
                           hA, hW, hC, 0, nullptr, nullptr);
    };

    // ---- input projection + relu ----
    for (int t = 0; t < 3; ++t) {
        const float* bw = (const float*)d_in[LIN_IN + SPOS[t] * 2 + 0];
        const float* ww = (const float*)d_in[LIN_IN + SPOS[t] * 2 + 1];
        gemm4(x_in[t], FIN[t], ww, HID, bw, x_a + (size_t)OFFN[t] * HID, HID,
              NN[t], FIN[t], HID, 1, nullptr, nullptr);
    }

    float* xc = x_a; float* xn = x_b;
    for (int layer = 0; layer < 2; ++layer) {
        const int cb = CONV0 + layer * CONV_STRIDE;

        // fused KQV projection per node type: [N,64] @ [64,192] + b
        for (int t = 0; t < 3; ++t) {
            const float* bw = (const float*)d_in[cb + 6 + SPOS[t] * 2 + 0];
            const float* ww = (const float*)d_in[cb + 6 + SPOS[t] * 2 + 1];
            gemm4(xc + (size_t)OFFN[t] * HID, HID, ww, 192, bw,
                  kqv + (size_t)OFFN[t] * 192, 192, NN[t], HID, 192, 0, nullptr, nullptr);
        }

        hipLaunchKernelGGL(fill_f32, dim3(2048), dim3(256), 0, stream, mmax, -INFINITY, (long long)NTOT_NODES * 4);
        hipLaunchKernelGGL(fill_f32, dim3(2048), dim3(256), 0, stream, den, 0.f, (long long)NTOT_NODES * 4);
        hipLaunchKernelGGL(fill_f32, dim3(2048), dim3(256), 0, stream, agg, 0.f, (long long)NTOT_NODES * HID);

        // pass 1: kt = K @ k_rel[h] (WMMA, heads in grid.z), then logits + segment max
        for (int j = 0; j < 6; ++j) {
            const EM& m_ = em[j];
            const float* krel = (const float*)d_in[cb + 0  + m_.sidx];
            const float* prel = (const float*)d_in[cb + 18 + m_.sidx];
            gemm1h(kqv + (size_t)m_.offs * 192 + 0, 192, krel, DH, nullptr,
                   trans, HID, m_.Ns, DH, DH, DH, DH * DH, DH, NHEAD);
            const int* src = edg[m_.ei];
            const int* dst = edg[m_.ei] + m_.E;
            int nthr = m_.E * 4;
            hipLaunchKernelGGL(edge_logits, dim3((nthr + 255) / 256), dim3(256), 0, stream,
                               src, dst, m_.E, trans, kqv + (size_t)m_.offd * 192, prel,
                               logits + (size_t)m_.eoff * 4, mmax, m_.offd);
        }

        // pass 2: exp + denominator
        for (int j = 0; j < 6; ++j) {
            const EM& m_ = em[j];
            const int* dst = edg[m_.ei] + m_.E;
            int nthr = m_.E * 4;
            hipLaunchKernelGGL(edge_exp_den, dim3((nthr + 255) / 256), dim3(256), 0, stream,
                               dst, m_.E, logits + (size_t)m_.eoff * 4, mmax, den, m_.offd);
        }

        // pass 3: vt = V @ v_rel[h] (WMMA), then alpha-weighted scatter-add
        for (int j = 0; j < 6; ++j) {
            const EM& m_ = em[j];
            const float* vrel = (const float*)d_in[cb + 27 + m_.sidx];
            gemm1h(kqv + (size_t)m_.offs * 192 + 128, 192, vrel, DH, nullptr,
                   trans, HID, m_.Ns, DH, DH, DH, DH * DH, DH, NHEAD);
            const int* src = edg[m_.ei];
            const int* dst = edg[m_.ei] + m_.E;
            int nthr = m_.E * 4;
            hipLaunchKernelGGL(edge_aggregate, dim3((nthr + 255) / 256), dim3(256), 0, stream,
                               src, dst, m_.E, logits + (size_t)m_.eoff * 4, den, trans, agg, m_.offd);
        }

        // out projection: gelu(agg) @ W + b, sigmoid-skip blend with xc
        hipLaunchKernelGGL(gelu_fwd, dim3(2048), dim3(256), 0, stream,
                           agg, gact, (long long)NTOT_NODES * HID);
        for (int t = 0; t < 3; ++t) {
            const float* bw = (const float*)d_in[cb + 12 + SPOS[t] * 2 + 0];
            const float* ww = (const float*)d_in[cb + 12 + SPOS[t] * 2 + 1];
            const float* sk = (const float*)d_in[cb + 24 + SPOS[t]];
            gemm4(gact + (size_t)OFFN[t] * HID, HID, ww, HID, bw,
                  xn + (size_t)OFFN[t] * HID, HID, NN[t], HID, HID, 0,
                  xc + (size_t)OFFN[t] * HID, sk);
        }
        float* tmp = xc; xc = xn; xn = tmp;
    }

    // final head: x_player @ [64,8] + b  (clamped N=8 tile)
    {
        dim3 g((unsigned)((NN[0] + MROWS - 1) / MROWS), 1, 1);
        hipLaunchKernelGGL((gemm_wmma<1>), g, dim3(256), 0, stream,
                           xc, HID, (const float*)d_in[LIN_OUT_W], 8,
                           (const float*)d_in[LIN_OUT_B], (float*)d_out, 8,
                           NN[0], HID, 8, 0L, 0L, 0L, 0, nullptr, nullptr);
    }
}